// MapNMSFreeCoder_24464133718516
// MI455X (gfx1250) — compile-verified
//
#include <hip/hip_runtime.h>
#include <hip/hip_bf16.h>

// ---------------- problem constants (from reference) ----------------
#define QN     100000
#define CN     10
#define NN     (QN * CN)          // 1,000,000 scores per batch
#define BATCH  8
#define KMAX   100
#define NPTS   20
#define CAP    512                // candidate buffer per batch (KMAX + tie slack)

// streaming kernels
#define STPB   256                // 8 wave32 waves
#define NBLK   128                // blocks per batch -> 1024 blocks total
#define TILE_S (STPB * 4)         // 1024 floats per staged tile (b128/lane)
#define NTILE_FULL (NN / TILE_S)  // 976 full tiles
#define TAIL_START (NTILE_FULL * TILE_S)  // 999424

// final kernel
#define FTPB   512

// output section offsets (floats), tuple order: boxes,scores,labels,pts,mask
#define OFF_SCR (BATCH * KMAX * 4)                   // 3200
#define OFF_LBL (OFF_SCR + BATCH * KMAX)             // 4000
#define OFF_PTS (OFF_LBL + BATCH * KMAX)             // 4800
#define OFF_MSK (OFF_PTS + BATCH * KMAX * NPTS * 2)  // 36800

// denorm constants: SX = 15-(-15)=30, SY = 30-(-30)=60
#define SXC 30.0f
#define SYC 60.0f
#define OXC (-15.0f)
#define OYC (-30.0f)
#define SCORE_THR 0.1f

// workspace layout (bytes)
#define WS_HIST_OFF  0                               // 8*256 u32 = 8192 B
#define WS_STATE_OFF (BATCH * 256 * 4)               // 8*4 u32 (prefix,pmask,krem,candcnt)
#define WS_CAND_OFF  (WS_STATE_OFF + BATCH * 4 * 4)  // 8*512 u64 = 32768 B

// -------- CDNA5 async global->LDS path (probe-confirmed lowering) --------
#if defined(__has_builtin)
#if __has_builtin(__builtin_amdgcn_global_load_async_to_lds_b128) && \
    __has_builtin(__builtin_amdgcn_s_wait_asynccnt)
#define USE_ASYNC_LDS 1
#endif
#endif
#ifndef USE_ASYNC_LDS
#define USE_ASYNC_LDS 0
#endif

#if USE_ASYNC_LDS
typedef int v4i __attribute__((vector_size(4 * sizeof(int))));
__device__ __forceinline__ void async_issue16(const float* gp, float* slot) {
  __builtin_amdgcn_global_load_async_to_lds_b128((v4i*)gp, (v4i*)slot, 0, 0);
}
#endif

// monotonic float<->uint key mapping (order-preserving)
__device__ __forceinline__ unsigned f2key(float f) {
  unsigned u = __float_as_uint(f);
  return u ^ ((u & 0x80000000u) ? 0xFFFFFFFFu : 0x80000000u);
}
__device__ __forceinline__ float key2f(unsigned k) {
  unsigned u = (k & 0x80000000u) ? (k ^ 0x80000000u) : ~k;
  return __uint_as_float(u);
}
__device__ __forceinline__ float sigmoidf(float x) {
  return 1.0f / (1.0f + expf(-x));
}

// ---------------- kernel 0: init workspace ----------------
__global__ void init_kernel(unsigned* __restrict__ hist, unsigned* __restrict__ state) {
  const int tid = threadIdx.x + blockIdx.x * blockDim.x;
  if (tid < BATCH * 256) hist[tid] = 0u;
  if (tid < BATCH) {
    state[tid * 4 + 0] = 0u;      // prefix
    state[tid * 4 + 1] = 0u;      // pmask
    state[tid * 4 + 2] = KMAX;    // krem (1-based rank remaining)
    state[tid * 4 + 3] = 0u;      // candidate count
  }
}

// ---------------- kernel 1 (x4): device-wide histogram of one radix digit ----------------
__global__ __launch_bounds__(STPB)
void hist_kernel(const float* __restrict__ cls, unsigned* __restrict__ hist,
                 const unsigned* __restrict__ state, int shift) {
  const int b   = blockIdx.y;
  const int blk = blockIdx.x;
  const int tid = threadIdx.x;
  const float* s = cls + (size_t)b * NN;

  __shared__ unsigned lh[256];
  __shared__ float tile[2][TILE_S];
  for (int i = tid; i < 256; i += STPB) lh[i] = 0u;
  __syncthreads();

  const unsigned prefix = state[b * 4 + 0];
  const unsigned pmask  = state[b * 4 + 1];

#if USE_ASYNC_LDS
  // double-buffered async global->LDS staging: keep one tile in flight,
  // wait asynccnt<=1 so the *older* transfer is complete (in-order per wave)
  int parity = 0;
  if (blk < NTILE_FULL)
    async_issue16(s + blk * TILE_S + tid * 4, &tile[0][tid * 4]);
  for (int t = blk; t < NTILE_FULL; t += NBLK) {
    const int tn = t + NBLK;
    if (tn < NTILE_FULL) {
      async_issue16(s + tn * TILE_S + tid * 4, &tile[parity ^ 1][tid * 4]);
      __builtin_amdgcn_s_wait_asynccnt(1);
    } else {
      __builtin_amdgcn_s_wait_asynccnt(0);
    }
    const float* sl = &tile[parity][tid * 4];
    const float v0 = sl[0], v1 = sl[1], v2 = sl[2], v3 = sl[3];
#else
  for (int t = blk; t < NTILE_FULL; t += NBLK) {
    const float4 q = *reinterpret_cast<const float4*>(s + t * TILE_S + tid * 4);
    const float v0 = q.x, v1 = q.y, v2 = q.z, v3 = q.w;
#endif
    unsigned k0 = f2key(sigmoidf(v0));
    unsigned k1 = f2key(sigmoidf(v1));
    unsigned k2 = f2key(sigmoidf(v2));
    unsigned k3 = f2key(sigmoidf(v3));
    if ((k0 & pmask) == prefix) atomicAdd(&lh[(k0 >> shift) & 0xFFu], 1u);
    if ((k1 & pmask) == prefix) atomicAdd(&lh[(k1 >> shift) & 0xFFu], 1u);
    if ((k2 & pmask) == prefix) atomicAdd(&lh[(k2 >> shift) & 0xFFu], 1u);
    if ((k3 & pmask) == prefix) atomicAdd(&lh[(k3 >> shift) & 0xFFu], 1u);
#if USE_ASYNC_LDS
    parity ^= 1;
#endif
  }
  for (int i = TAIL_START + blk * STPB + tid; i < NN; i += NBLK * STPB) {
    unsigned k = f2key(sigmoidf(s[i]));
    if ((k & pmask) == prefix) atomicAdd(&lh[(k >> shift) & 0xFFu], 1u);
  }
  __syncthreads();
  for (int i = tid; i < 256; i += STPB)
    if (lh[i]) atomicAdd(&hist[b * 256 + i], lh[i]);
}

// ---------------- kernel 2 (x4): pick digit, update prefix state, re-zero hist ----------------
__global__ __launch_bounds__(256)
void select_kernel(unsigned* __restrict__ hist, unsigned* __restrict__ state, int shift) {
  const int b = blockIdx.x;
  const int tid = threadIdx.x;
  if (tid == 0) {
    const unsigned prefix = state[b * 4 + 0];
    const unsigned pmask  = state[b * 4 + 1];
    const unsigned krem   = state[b * 4 + 2];
    unsigned cum = 0, d = 0;
    for (int dd = 255; dd >= 0; --dd) {
      unsigned c = hist[b * 256 + dd];
      if (cum + c >= krem) { d = (unsigned)dd; state[b * 4 + 2] = krem - cum; break; }
      cum += c;
    }
    state[b * 4 + 0] = prefix | (d << shift);
    state[b * 4 + 1] = pmask  | (0xFFu << shift);
  }
  __syncthreads();
  hist[b * 256 + tid] = 0u;   // ready for next pass
}

// ---------------- kernel 3: collect all elements with key >= kth ----------------
__global__ __launch_bounds__(STPB)
void collect_kernel(const float* __restrict__ cls, unsigned* __restrict__ state,
                    unsigned long long* __restrict__ cand) {
  const int b   = blockIdx.y;
  const int blk = blockIdx.x;
  const int tid = threadIdx.x;
  const float* s = cls + (size_t)b * NN;
  __shared__ float tile[2][TILE_S];

  const unsigned kth = state[b * 4 + 0];   // exact key of the 100th-largest value

#if USE_ASYNC_LDS
  int parity = 0;
  if (blk < NTILE_FULL)
    async_issue16(s + blk * TILE_S + tid * 4, &tile[0][tid * 4]);
  for (int t = blk; t < NTILE_FULL; t += NBLK) {
    const int tn = t + NBLK;
    if (tn < NTILE_FULL) {
      async_issue16(s + tn * TILE_S + tid * 4, &tile[parity ^ 1][tid * 4]);
      __builtin_amdgcn_s_wait_asynccnt(1);
    } else {
      __builtin_amdgcn_s_wait_asynccnt(0);
    }
    const float* sl = &tile[parity][tid * 4];
    float v[4] = { sl[0], sl[1], sl[2], sl[3] };
#else
  for (int t = blk; t < NTILE_FULL; t += NBLK) {
    const float* gp = s + t * TILE_S + tid * 4;
    if (t + NBLK < NTILE_FULL) __builtin_prefetch(gp + NBLK * TILE_S, 0, 1);
    const float4 q = *reinterpret_cast<const float4*>(gp);
    float v[4] = { q.x, q.y, q.z, q.w };
#endif
    #pragma unroll
    for (int k = 0; k < 4; ++k) {
      unsigned key = f2key(sigmoidf(v[k]));
      if (key >= kth) {
        unsigned p = atomicAdd(&state[b * 4 + 3], 1u);
        unsigned idx = (unsigned)(t * TILE_S + tid * 4 + k);
        if (p < CAP)   // pack: key desc primary, index asc secondary (via ~index)
          cand[b * CAP + p] = ((unsigned long long)key << 32) |
                              (unsigned long long)(0xFFFFFFFFu - idx);
      }
    }
#if USE_ASYNC_LDS
    parity ^= 1;
#endif
  }
  for (int i = TAIL_START + blk * STPB + tid; i < NN; i += NBLK * STPB) {
    unsigned key = f2key(sigmoidf(s[i]));
    if (key >= kth) {
      unsigned p = atomicAdd(&state[b * 4 + 3], 1u);
      if (p < CAP)
        cand[b * CAP + p] = ((unsigned long long)key << 32) |
                            (unsigned long long)(0xFFFFFFFFu - (unsigned)i);
    }
  }
}

// ---------------- kernel 4: sort candidates, decode, mask, write ----------------
__global__ __launch_bounds__(FTPB)
void finalize_kernel(const float* __restrict__ bbox, const float* __restrict__ pts,
                     const unsigned* __restrict__ state,
                     const unsigned long long* __restrict__ cand,
                     float* __restrict__ out) {
  const int b = blockIdx.x;
  const int tid = threadIdx.x;
  __shared__ unsigned long long cbuf[CAP];

  const unsigned cnt = min(state[b * 4 + 3], (unsigned)CAP);
  cbuf[tid] = (tid < (int)cnt) ? cand[b * CAP + tid] : 0ull;
  __syncthreads();

  // bitonic sort, descending on packed (key, ~index) -> deterministic output
  for (unsigned ksz = 2; ksz <= CAP; ksz <<= 1) {
    for (unsigned j = ksz >> 1; j > 0; j >>= 1) {
      unsigned i = tid;
      unsigned ixj = i ^ j;
      if (ixj > i) {
        unsigned long long a = cbuf[i], c = cbuf[ixj];
        bool descSeg = ((i & ksz) == 0);
        if (descSeg ? (a < c) : (a > c)) { cbuf[i] = c; cbuf[ixj] = a; }
      }
      __syncthreads();
    }
  }

  const float smax = key2f((unsigned)(cbuf[0] >> 32));
  if (tid < KMAX) {
    const unsigned long long sv = cbuf[tid];
    const unsigned key  = (unsigned)(sv >> 32);
    const unsigned fidx = 0xFFFFFFFFu - (unsigned)(sv & 0xFFFFFFFFull);
    const float score   = key2f(key);
    const int label = (int)(fidx % CN);
    const int qi    = (int)(fidx / CN);

    // denorm bbox: cxcywh -> xyxy, scale (30,60), offset (-15,-30)
    const float* bb = bbox + ((size_t)b * QN + qi) * 4;
    const float cx = bb[0], cy = bb[1], w = bb[2], h = bb[3];
    const float x1 = (cx - 0.5f * w) * SXC + OXC;
    const float y1 = (cy - 0.5f * h) * SYC + OYC;
    const float x2 = (cx + 0.5f * w) * SXC + OXC;
    const float y2 = (cy + 0.5f * h) * SYC + OYC;

    const bool rmask = (x1 >= -20.f) && (y1 >= -35.f) && (x2 >= -20.f) && (y2 >= -35.f) &&
                       (x1 <=  20.f) && (y1 <=  35.f) && (x2 <=  20.f) && (y2 <=  35.f);

    // threshold mask with the reference's relaxation-loop semantics
    bool tmask;
    if (smax > SCORE_THR) {
      tmask = (score > SCORE_THR);
    } else {
      float tmp = SCORE_THR;
      for (;;) {
        tmp *= 0.9f;
        if (tmp < 0.01f) { tmask = (score > -1.0f); break; }
        if (smax >= tmp) { tmask = (score >= tmp);  break; }
      }
    }
    const bool keep = rmask && tmask;

    const int base = b * KMAX + tid;
    float* oBox = out;
    float* oScr = out + OFF_SCR;
    float* oLbl = out + OFF_LBL;
    float* oPts = out + OFF_PTS;
    float* oMsk = out + OFF_MSK;

    oBox[base * 4 + 0] = keep ? x1 : 0.f;
    oBox[base * 4 + 1] = keep ? y1 : 0.f;
    oBox[base * 4 + 2] = keep ? x2 : 0.f;
    oBox[base * 4 + 3] = keep ? y2 : 0.f;
    oScr[base] = keep ? score : 0.f;
    oLbl[base] = keep ? (float)label : 0.f;
    oMsk[base] = keep ? 1.0f : 0.0f;

    const float* pp = pts + ((size_t)b * QN + qi) * (NPTS * 2);
    float* op = oPts + (size_t)base * (NPTS * 2);
    #pragma unroll
    for (int j = 0; j < NPTS; ++j) {
      const float px = pp[2 * j + 0];
      const float py = pp[2 * j + 1];
      op[2 * j + 0] = keep ? (px * SXC + OXC) : 0.f;
      op[2 * j + 1] = keep ? (py * SYC + OYC) : 0.f;
    }
  }
}

extern "C" void kernel_launch(void* const* d_in, const int* in_sizes, int n_in,
                              void* d_out, int out_size, void* d_ws, size_t ws_size,
                              hipStream_t stream) {
  (void)in_sizes; (void)n_in; (void)out_size; (void)ws_size;
  const float* cls  = (const float*)d_in[0];   // [8,100000,10]
  const float* bbox = (const float*)d_in[1];   // [8,100000,4]
  const float* pts  = (const float*)d_in[2];   // [8,100000,20,2]
  float* out = (float*)d_out;                  // 37600 floats

  unsigned* hist  = (unsigned*)((char*)d_ws + WS_HIST_OFF);
  unsigned* state = (unsigned*)((char*)d_ws + WS_STATE_OFF);
  unsigned long long* cand = (unsigned long long*)((char*)d_ws + WS_CAND_OFF);

  init_kernel<<<dim3((BATCH * 256 + 255) / 256), dim3(256), 0, stream>>>(hist, state);

  const dim3 sgrid(NBLK, BATCH);
  for (int pass = 0; pass < 4; ++pass) {
    const int shift = 24 - 8 * pass;
    hist_kernel<<<sgrid, dim3(STPB), 0, stream>>>(cls, hist, state, shift);
    select_kernel<<<dim3(BATCH), dim3(256), 0, stream>>>(hist, state, shift);
  }
  collect_kernel<<<sgrid, dim3(STPB), 0, stream>>>(cls, state, cand);
  finalize_kernel<<<dim3(BATCH), dim3(FTPB), 0, stream>>>(bbox, pts, state, cand, out);
}